// CompressAttn_31155692765754
// MI455X (gfx1250) — compile-verified
//
#include <hip/hip_runtime.h>

// CDNA5 / gfx1250. Memory-bound strided window reduction:
//   y[b,m,h,d] = sum_{k<32} w[k] * x[b, 16m+k, h, d] + bias[d]
// U/V half-kernel split (stride = K/2):
//   U[j,d] = sum_{r<16} w[r]    * x[16j+r,d]
//   V[j,d] = sum_{r<16} w[16+r] * x[16j+r,d]
//   y[m,d] = U[m,d] + V[m+1,d] + bias[d]
// One wave owns a 16-wide d tile; per j: 16x16 f32 x-block is staged into
// LDS with GLOBAL_LOAD_ASYNC_TO_LDS_B128 (double buffered, ASYNCcnt), then
// consumed by 4 chained V_WMMA_F32_16X16X4_F32 (rows 0/1 of C = U/V).

typedef __attribute__((ext_vector_type(2))) float v2f;
typedef __attribute__((ext_vector_type(8))) float v8f;

#define B_  4
#define N_  16384
#define H_  4
#define D_  128
#define K_  32
#define M_  1023          // (N - K)/16 + 1
#define RS_ (H_ * D_)     // 512 floats between consecutive n rows (fixed b,h)

// ---------------------------------------------------------------- bias ----
__global__ __launch_bounds__(128)
void bias_kernel(const float* __restrict__ w, const float* __restrict__ pe,
                 float* __restrict__ bias)
{
    const int d = threadIdx.x;            // 0..127
    float s = 0.f;
#pragma unroll
    for (int k = 0; k < K_; ++k) s += w[k] * pe[k * D_ + d];
    bias[d] = s;
}

// ---------------------------------------------------------------- main ----
__global__ __launch_bounds__(256)
void compress_conv_wmma(const float* __restrict__ x,
                        const float* __restrict__ w,
                        const float* __restrict__ bias,
                        float* __restrict__ y)
{
    // 8 waves x (2 buffers x 16 rows x 16 floats)
    __shared__ float sx[8][2][256];

    const int lane = threadIdx.x & 31;
    const int wave = threadIdx.x >> 5;   // 0..7 -> d tile
    const int d0   = wave * 16;
    const int col  = lane & 15;

    const int blk = blockIdx.x;          // 1024 blocks = 16 (b,h) * 64 m-tiles
    const int bh  = blk >> 6;
    const int mt  = blk & 63;
    const int b   = bh >> 2;
    const int h   = bh & 3;

    const long xbase = (long)b * N_ * RS_ + (long)h * D_;

    // ---- A matrix (weights), 16x4 f32 per K chunk: branch-free setup ----
    // lanes 0-15: M=lane, K={0,1}; lanes 16-31: M=lane-16, K={2,3}
    // Row M=0 = w[0:16] (U), row M=1 = w[16:32] (V), other rows 0.
    const int Mrow  = lane & 15;
    const int kbase = (lane < 16) ? 0 : 2;
    float aw[4][2];
#pragma unroll
    for (int c = 0; c < 4; ++c)
#pragma unroll
        for (int v = 0; v < 2; ++v) {
            const int k = 4 * c + kbase + v;
            float wa = w[k];
            float wb = w[16 + k];
            // Opaque barrier: keep the loads unconditional so the selects
            // lower to v_cndmask instead of exec-masked branch blocks.
            asm volatile("" : "+v"(wa), "+v"(wb));
            aw[c][v] = (Mrow == 0) ? wa : ((Mrow == 1) ? wb : 0.f);
        }

    const float bias_d = bias[d0 + col];  // precomputed by bias_kernel

    // ---- async staging addresses -----------------------------------------
    // Per j: 16 rows x 64B. 2x B128 instrs: lane L -> row = (i*32+L)>>2,
    // 16B quadrant q = L&3. Low 32 bits of a generic shared pointer are the
    // wave-relative LDS byte offset required by async VDST.
    const int row0 = lane >> 2;           // 0..7   (i=0); i=1 adds 8
    const int q4   = (lane & 3) * 4;      // float offset of 16B quadrant
    const unsigned lds0 =
        (unsigned)(uintptr_t)(&sx[wave][0][row0 * 16 + q4]);
    const float* gbase = x + xbase + (long)row0 * RS_ + d0 + q4;

    const int j0   = mt * 16;
    const int jmax = (1023 - j0 < 16) ? (1023 - j0) : 16;  // inclusive jj bound

    auto issue_async = [&](int j, int buf) {
        const float* ga = gbase + (long)(16 * j) * RS_;
        const unsigned la = lds0 + (unsigned)buf * 1024u;
        asm volatile("global_load_async_to_lds_b128 %0, %1, off"
                     :: "v"(la), "v"(ga) : "memory");
        asm volatile("global_load_async_to_lds_b128 %0, %1, off"
                     :: "v"(la + 512u), "v"(ga + 8 * RS_) : "memory");
    };

    issue_async(j0, 0);

    float prevU = 0.f;
#pragma unroll 1
    for (int jj = 0; jj <= jmax; ++jj) {
        const int j   = j0 + jj;
        const int buf = jj & 1;

        if (jj < jmax) {
            issue_async(j + 1, buf ^ 1);
            // allow the 2 just-issued ops to stay in flight
            asm volatile("s_wait_asynccnt 0x2" ::: "memory");
        } else {
            asm volatile("s_wait_asynccnt 0x0" ::: "memory");
        }

        const float* s = &sx[wave][buf][0];
        v8f acc = {0.f, 0.f, 0.f, 0.f, 0.f, 0.f, 0.f, 0.f};
#pragma unroll
        for (int ch = 0; ch < 4; ++ch) {
            v2f a;
            a[0] = aw[ch][0];
            a[1] = aw[ch][1];
            // B 4x16: lane = column d; elems -> K rows {0,1}/{2,3}
            const int r0 = 4 * ch + kbase;
            v2f bm;
            bm[0] = s[r0 * 16 + col];           // paired ds reads, no conflicts
            bm[1] = s[(r0 + 1) * 16 + col];
            acc = __builtin_amdgcn_wmma_f32_16x16x4_f32(
                false, a, false, bm, (short)0, acc, false, false);
        }

        // D rows: M=0 -> acc[0] lanes 0-15 (=U[j]); M=1 -> acc[1] (=V[j])
        if (jj > 0 && lane < 16) {
            const int m = j - 1;                // 0..1022
            y[(((long)b * M_ + m) * H_ + h) * D_ + d0 + col] =
                prevU + acc[1] + bias_d;
        }
        prevU = acc[0];
    }
}

extern "C" void kernel_launch(void* const* d_in, const int* in_sizes, int n_in,
                              void* d_out, int out_size, void* d_ws, size_t ws_size,
                              hipStream_t stream) {
    const float* x  = (const float*)d_in[0];
    const float* w  = (const float*)d_in[1];
    const float* pe = (const float*)d_in[2];
    // d_in[3] is the python scalar stride (=16), baked into the kernel.
    float* y    = (float*)d_out;
    float* bias = (float*)d_ws;   // 128 floats of scratch

    bias_kernel<<<dim3(1), dim3(128), 0, stream>>>(w, pe, bias);
    compress_conv_wmma<<<dim3(16 * 64), dim3(256), 0, stream>>>(x, w, bias, y);
}